// GAT_edge_2_41137196761626
// MI455X (gfx1250) — compile-verified
//
#include <hip/hip_runtime.h>
#include <hip/hip_bf16.h>
#include <math.h>

// ---------------------------------------------------------------------------
// GAT (4-layer, edge-featured) for MI455X / gfx1250, wave32.
// Dense GEMMs use V_WMMA_F32_16X16X4_F32 (full fp32 matrix pipe).
// ---------------------------------------------------------------------------

typedef __attribute__((ext_vector_type(2))) float v2f;
typedef __attribute__((ext_vector_type(8))) float v8f;

#define NN 32768
#define EE 524288
#define DD 128
#define NG 64

// ---- monotonic float<->uint encoding for atomicMax on floats ---------------
__device__ __forceinline__ unsigned fenc(float f) {
  unsigned u = __float_as_uint(f);
  return (u & 0x80000000u) ? ~u : (u | 0x80000000u);
}
__device__ __forceinline__ float fdec(unsigned u) {
  return __uint_as_float((u & 0x80000000u) ? (u & 0x7FFFFFFFu) : ~u);
}

// ---------------------------------------------------------------------------
// GEMM: C[M,128] = A[M,128] * B[128,128], fp32 WMMA 16x16x4.
// One wave -> 16 rows x 128 cols (8 v8f accumulators). B staged in LDS with
// stride 136 so half-wave row pairs (k, k+2) land in disjoint bank groups.
// block = 256 threads (8 waves), grid = M/128 blocks. M must be mult of 128.
// ---------------------------------------------------------------------------
__global__ __launch_bounds__(256) void gemm_f32_k128(
    const float* __restrict__ A, const float* __restrict__ B,
    float* __restrict__ C, int M) {
  __shared__ float Bs[128 * 136];
  const int tid = threadIdx.x;
  for (int i = tid; i < 128 * 128; i += 256) {
    int r = i >> 7, c = i & 127;
    Bs[r * 136 + c] = B[i];
  }
  __syncthreads();

  const int wave = (blockIdx.x * 256 + tid) >> 5;
  const int lane = tid & 31;
  const int m0 = wave * 16;
  if (m0 >= M) return;  // wave-uniform

  const int row = lane & 15;     // A row within tile / B-D column index
  const int kh  = lane >> 4;     // 0: K=k,k+1   1: K=k+2,k+3

  v8f acc[8] = {};
  const float* Arow = A + (size_t)(m0 + row) * 128;
#pragma unroll 4
  for (int k = 0; k < 128; k += 4) {
    v2f a;
    a.x = Arow[k + 2 * kh + 0];
    a.y = Arow[k + 2 * kh + 1];
#pragma unroll
    for (int t = 0; t < 8; ++t) {
      v2f b;
      b.x = Bs[(k + 2 * kh + 0) * 136 + t * 16 + row];
      b.y = Bs[(k + 2 * kh + 1) * 136 + t * 16 + row];
      acc[t] = __builtin_amdgcn_wmma_f32_16x16x4_f32(
          false, a, false, b, (short)0, acc[t], false, false);
    }
  }
  // D layout: vgpr v -> M = m0+v (lanes 0-15) / m0+v+8 (lanes 16-31), N = lane&15
  const int mr = (lane < 16) ? 0 : 8;
#pragma unroll
  for (int t = 0; t < 8; ++t)
#pragma unroll
    for (int v = 0; v < 8; ++v)
      C[(size_t)(m0 + mr + v) * 128 + t * 16 + row] = acc[t][v];
}

// ---- self-loop attr: segment mean of incoming edge_attr --------------------
__global__ void loop_accum(const float* __restrict__ ea, const int* __restrict__ dst,
                           float* __restrict__ lsum, float* __restrict__ cnt) {
  int e = blockIdx.x, f = threadIdx.x;
  int d = dst[e];
  if (f == 0) atomicAdd(&cnt[d], 1.0f);
  atomicAdd(&lsum[(size_t)d * DD + f], ea[(size_t)e * DD + f]);
}
__global__ void loop_finish(float* __restrict__ la, const float* __restrict__ cnt) {
  int n = blockIdx.x, f = threadIdx.x;
  la[(size_t)n * DD + f] /= fmaxf(cnt[n], 1.0f);
}

// ---- folded edge-attention matrix M[d,h] = sum_c W_e[d,hC+c]*att_e[h,c] ----
__global__ void make_Me(const float* __restrict__ We, const float* __restrict__ atte,
                        float* __restrict__ Me, int H, int Cc) {
  int dI = threadIdx.x;  // 128 threads
  for (int h = 0; h < H; ++h) {
    float s = 0.f;
    for (int c = 0; c < Cc; ++c) s += We[dI * 128 + h * Cc + c] * atte[h * Cc + c];
    Me[dI * H + h] = s;
  }
}

// ---- per-node attention scalars a_src/a_dst from hh ------------------------
__global__ void node_attn(const float* __restrict__ hh, const float* __restrict__ as,
                          const float* __restrict__ ad, float* __restrict__ a_src,
                          float* __restrict__ a_dst, int H, int Cc) {
  int idx = blockIdx.x * blockDim.x + threadIdx.x;
  if (idx >= NN * H) return;
  int n = idx / H, h = idx - n * H;
  const float* row = hh + (size_t)n * 128 + h * Cc;
  float s = 0.f, d = 0.f;
  for (int c = 0; c < Cc; ++c) { float v = row[c]; s += v * as[h * Cc + c]; d += v * ad[h * Cc + c]; }
  a_src[idx] = s; a_dst[idx] = d;
}

__global__ void amax_init(unsigned* __restrict__ a, int n) {
  int i = blockIdx.x * blockDim.x + threadIdx.x;
  if (i < n) a[i] = 0x007FFFFFu;  // fenc(-inf)
}

// ---- pass 1: alpha = leaky_relu(a_src+a_dst+a_e), segment max via atomics --
__global__ __launch_bounds__(256) void edge_alpha(
    const float* __restrict__ ea, const float* __restrict__ la,
    const int* __restrict__ srcA, const int* __restrict__ dstA,
    const float* __restrict__ Me, const float* __restrict__ a_src,
    const float* __restrict__ a_dst, float* __restrict__ alpha,
    unsigned* __restrict__ amax, int H) {
  int wave = (blockIdx.x * 256 + threadIdx.x) >> 5;
  int lane = threadIdx.x & 31;
  if (wave >= EE + NN) return;
  int e = wave, s, d;
  const float* row;
  if (e < EE) { s = srcA[e]; d = dstA[e]; row = ea + (size_t)e * DD; }
  else        { s = d = e - EE;           row = la + (size_t)(e - EE) * DD; }
  float acc[8];
  for (int h = 0; h < 8; ++h) acc[h] = 0.f;
  for (int j = 0; j < 4; ++j) {
    int f = lane + 32 * j;
    float v = row[f];
    for (int h = 0; h < H; ++h) acc[h] += v * Me[f * H + h];
  }
  for (int h = 0; h < H; ++h)
    for (int m = 16; m >= 1; m >>= 1) acc[h] += __shfl_xor(acc[h], m, 32);
  if (lane < H) {
    float al = a_src[s * H + lane] + a_dst[d * H + lane] + acc[lane];
    al = (al > 0.f) ? al : 0.2f * al;
    alpha[(size_t)e * H + lane] = al;
    atomicMax(&amax[d * H + lane], fenc(al));
  }
}

// ---- pass 2: ex = exp(alpha - amax[dst]); den[dst] += ex -------------------
__global__ void edge_exp(const int* __restrict__ dstA, float* __restrict__ alpha,
                         const unsigned* __restrict__ amax, float* __restrict__ den,
                         int H) {
  int idx = blockIdx.x * blockDim.x + threadIdx.x;
  int tot = (EE + NN) * H;
  if (idx >= tot) return;
  int e = idx / H, h = idx - e * H;
  int d = (e < EE) ? dstA[e] : e - EE;
  float ex = expf(alpha[idx] - fdec(amax[d * H + h]));
  alpha[idx] = ex;
  atomicAdd(&den[d * H + h], ex);
}

// ---- pass 3: out[dst] += hh[src] * ex/(den+eps) ----------------------------
__global__ __launch_bounds__(256) void edge_aggregate(
    const float* __restrict__ hh, const int* __restrict__ srcA,
    const int* __restrict__ dstA, const float* __restrict__ exv,
    const float* __restrict__ den, float* __restrict__ outa, int H, int Cc) {
  int wave = (blockIdx.x * 256 + threadIdx.x) >> 5;
  int lane = threadIdx.x & 31;
  if (wave >= EE + NN) return;
  int e = wave, s, d;
  if (e < EE) { s = srcA[e]; d = dstA[e]; } else s = d = e - EE;
  for (int j = 0; j < 4; ++j) {
    int f = lane + 32 * j;
    int h = f / Cc;
    float w = exv[(size_t)e * H + h] / (den[d * H + h] + 1e-16f);
    atomicAdd(&outa[(size_t)d * 128 + f], hh[(size_t)s * 128 + f] * w);
  }
}

// ---- BN column statistics: fp32 block partials -> fp64 atomics -------------
__global__ __launch_bounds__(128) void bn_stats(const float* __restrict__ X,
                                                double* __restrict__ dsum,
                                                double* __restrict__ dsq) {
  int f = threadIdx.x;
  int r0 = blockIdx.x * 128;
  float s = 0.f, q = 0.f;
  for (int r = r0; r < r0 + 128; ++r) {
    float v = X[(size_t)r * 128 + f];
    s += v; q += v * v;
  }
  atomicAdd(&dsum[f], (double)s);
  atomicAdd(&dsq[f], (double)q);
}

// ---- node epilogue: BN (bias cancels), LN, skip add, ELU (in place) --------
__global__ __launch_bounds__(128) void node_epilogue(
    float* __restrict__ X, const float* __restrict__ skip,
    const float* __restrict__ skip_b, const float* __restrict__ bn_g,
    const float* __restrict__ bn_b, const float* __restrict__ ln_g,
    const float* __restrict__ ln_b, const double* __restrict__ dsum,
    const double* __restrict__ dsq) {
  __shared__ float red[128];
  int n = blockIdx.x, f = threadIdx.x;
  float v = X[(size_t)n * 128 + f];
  float mean = (float)(dsum[f] * (1.0 / NN));
  float var = (float)(dsq[f] * (1.0 / NN)) - mean * mean;
  v = (v - mean) * rsqrtf(fmaxf(var, 0.f) + 1e-5f) * bn_g[f] + bn_b[f];
  red[f] = v; __syncthreads();
  for (int s = 64; s > 0; s >>= 1) { if (f < s) red[f] += red[f + s]; __syncthreads(); }
  float mu = red[0] * (1.f / 128.f); __syncthreads();
  float dv = v - mu;
  red[f] = dv * dv; __syncthreads();
  for (int s = 64; s > 0; s >>= 1) { if (f < s) red[f] += red[f + s]; __syncthreads(); }
  float var2 = red[0] * (1.f / 128.f);
  v = dv * rsqrtf(var2 + 1e-5f) * ln_g[f] + ln_b[f];
  if (skip) v += skip[(size_t)n * 128 + f] + skip_b[f];
  X[(size_t)n * 128 + f] = (v > 0.f) ? v : (expf(v) - 1.f);
}

// ---- global mean pool ------------------------------------------------------
__global__ void pool_accum(const float* __restrict__ h, const int* __restrict__ batch,
                           float* __restrict__ gsum, float* __restrict__ gcnt) {
  int n = blockIdx.x, f = threadIdx.x;
  int g = batch[n];
  atomicAdd(&gsum[(size_t)g * 128 + f], h[(size_t)n * 128 + f]);
  if (f == 0) atomicAdd(&gcnt[g], 1.0f);
}
__global__ void pool_finish(const float* __restrict__ gsum, const float* __restrict__ gcnt,
                            float* __restrict__ out) {
  int g = blockIdx.x, f = threadIdx.x;
  out[g * 128 + f] = gsum[g * 128 + f] / fmaxf(gcnt[g], 1.0f);
}

// ---------------------------------------------------------------------------
extern "C" void kernel_launch(void* const* d_in, const int* in_sizes, int n_in,
                              void* d_out, int out_size, void* d_ws, size_t ws_size,
                              hipStream_t stream) {
  (void)in_sizes; (void)n_in; (void)out_size; (void)ws_size;
  const float* x = (const float*)d_in[0];
  const int* ei = (const int*)d_in[1];
  const int* srcA = ei;
  const int* dstA = ei + EE;
  const float* edge_attr = (const float*)d_in[2];
  const int* batch = (const int*)d_in[3];

  // params in jax pytree (sorted-key) order
  struct Layer { const float *W,*We,*att_dst,*att_edge,*att_src,*bias,*bn_b,*bn_g,*ln_b,*ln_g,*skip_W,*skip_b; };
  Layer L[4];
  int p = 4;
  for (int i = 0; i < 4; ++i) {
    L[i].W        = (const float*)d_in[p++];
    L[i].We       = (const float*)d_in[p++];
    L[i].att_dst  = (const float*)d_in[p++];
    L[i].att_edge = (const float*)d_in[p++];
    L[i].att_src  = (const float*)d_in[p++];
    L[i].bias     = (const float*)d_in[p++];
    L[i].bn_b     = (const float*)d_in[p++];
    L[i].bn_g     = (const float*)d_in[p++];
    L[i].ln_b     = (const float*)d_in[p++];
    L[i].ln_g     = (const float*)d_in[p++];
    if (i > 0) { L[i].skip_W = (const float*)d_in[p++]; L[i].skip_b = (const float*)d_in[p++]; }
    else       { L[i].skip_W = nullptr;                 L[i].skip_b = nullptr; }
  }

  // workspace carve-out
  char* w = (char*)d_ws; size_t off = 0;
  auto carve = [&](size_t bytes) -> void* {
    void* r = w + off; off = (off + bytes + 255) & ~(size_t)255; return r;
  };
  float*    hA    = (float*)   carve((size_t)NN * 128 * 4);
  float*    hB    = (float*)   carve((size_t)NN * 128 * 4);
  float*    hh    = (float*)   carve((size_t)NN * 128 * 4);
  float*    skip  = (float*)   carve((size_t)NN * 128 * 4);
  float*    la    = (float*)   carve((size_t)NN * 128 * 4);   // loop attr (sum -> mean)
  float*    alpha = (float*)   carve((size_t)(EE + NN) * 8 * 4);
  float*    cnt   = (float*)   carve((size_t)NN * 4);
  float*    a_src = (float*)   carve((size_t)NN * 8 * 4);
  float*    a_dst = (float*)   carve((size_t)NN * 8 * 4);
  unsigned* amax  = (unsigned*)carve((size_t)NN * 8 * 4);
  float*    den   = (float*)   carve((size_t)NN * 8 * 4);
  float*    Me    = (float*)   carve(128 * 8 * 4);
  double*   dsum  = (double*)  carve(128 * 8);
  double*   dsq   = (double*)  carve(128 * 8);
  float*    gsum  = (float*)   carve((size_t)NG * 128 * 4);
  float*    gcnt  = (float*)   carve((size_t)NG * 4);

  // self-loop attributes: per-dst mean of incoming edge_attr
  hipMemsetAsync(la, 0, (size_t)NN * 128 * 4, stream);
  hipMemsetAsync(cnt, 0, (size_t)NN * 4, stream);
  loop_accum<<<EE, 128, 0, stream>>>(edge_attr, dstA, la, cnt);
  loop_finish<<<NN, 128, 0, stream>>>(la, cnt);

  const int TOT = EE + NN;  // 557056, divisible by 8
  const float* cur = x;
  float* nxt = hA;
  for (int i = 0; i < 4; ++i) {
    const int H = (i < 3) ? 8 : 1;
    const int Cc = (i < 3) ? 16 : 128;

    gemm_f32_k128<<<NN / 128, 256, 0, stream>>>(cur, L[i].W, hh, NN);
    if (i > 0) gemm_f32_k128<<<NN / 128, 256, 0, stream>>>(cur, L[i].skip_W, skip, NN);

    node_attn<<<(NN * H + 255) / 256, 256, 0, stream>>>(hh, L[i].att_src, L[i].att_dst,
                                                        a_src, a_dst, H, Cc);
    make_Me<<<1, 128, 0, stream>>>(L[i].We, L[i].att_edge, Me, H, Cc);
    amax_init<<<(NN * H + 255) / 256, 256, 0, stream>>>(amax, NN * H);
    hipMemsetAsync(den, 0, (size_t)NN * H * 4, stream);
    hipMemsetAsync(nxt, 0, (size_t)NN * 128 * 4, stream);

    edge_alpha<<<TOT / 8, 256, 0, stream>>>(edge_attr, la, srcA, dstA, Me,
                                            a_src, a_dst, alpha, amax, H);
    edge_exp<<<(TOT * H + 255) / 256, 256, 0, stream>>>(dstA, alpha, amax, den, H);
    edge_aggregate<<<TOT / 8, 256, 0, stream>>>(hh, srcA, dstA, alpha, den, nxt, H, Cc);

    hipMemsetAsync(dsum, 0, 128 * 8, stream);
    hipMemsetAsync(dsq, 0, 128 * 8, stream);
    bn_stats<<<NN / 128, 128, 0, stream>>>(nxt, dsum, dsq);
    node_epilogue<<<NN, 128, 0, stream>>>(nxt, (i > 0) ? skip : nullptr, L[i].skip_b,
                                          L[i].bn_g, L[i].bn_b, L[i].ln_g, L[i].ln_b,
                                          dsum, dsq);
    cur = nxt;
    nxt = (cur == hA) ? hB : hA;
  }

  hipMemsetAsync(gsum, 0, (size_t)NG * 128 * 4, stream);
  hipMemsetAsync(gcnt, 0, (size_t)NG * 4, stream);
  pool_accum<<<NN, 128, 0, stream>>>(cur, batch, gsum, gcnt);
  pool_finish<<<NG, 128, 0, stream>>>(gsum, gcnt, (float*)d_out);
}